// AGGNodeGraph_47090021433990
// MI455X (gfx1250) — compile-verified
//
#include <hip/hip_runtime.h>

typedef __attribute__((ext_vector_type(2))) float v2f;
typedef __attribute__((ext_vector_type(8))) float v8f;

#define N_NODES 50000
#define N_EDGES 800000
#define IN_FEATS 256
#define HIDDEN 256
#define EDGES_PER_WAVE 16

// ---------------------------------------------------------------------------
// Kernel A: t = node_feats @ W + b  via V_WMMA_F32_16X16X4_F32.
// Each wave computes one 16x16 tile of t. Writes t to workspace AND d_out
// (d_out is seeded with t; kernel B accumulates agg on top of it).
// ---------------------------------------------------------------------------
__global__ __launch_bounds__(256) void gemm_bias_wmma_kernel(
    const float* __restrict__ A,   // node_feats [N_NODES, IN_FEATS]
    const float* __restrict__ W,   // [IN_FEATS, HIDDEN]
    const float* __restrict__ bv,  // [HIDDEN]
    float* __restrict__ t_ws,      // [N_NODES, HIDDEN] scratch
    float* __restrict__ out)       // [N_NODES, HIDDEN]
{
  const int lane = threadIdx.x & 31;
  const int waveId = (blockIdx.x * blockDim.x + threadIdx.x) >> 5;
  const int numColTiles = HIDDEN / 16;          // 16
  const int rowTile = waveId / numColTiles;
  const int colTile = waveId - rowTile * numColTiles;
  if (rowTile >= N_NODES / 16) return;

  const int m0 = rowTile * 16;
  const int n0 = colTile * 16;
  const int half = lane >> 4;                   // 0: lanes 0-15, 1: lanes 16-31
  const int l15 = lane & 15;

  // A row for this lane (16x4 f32 A tile: lanes 0-15 carry K={k,k+1},
  // lanes 16-31 carry K={k+2,k+3} for row M = lane&15).
  const float* arow = A + (m0 + l15) * IN_FEATS;
  // B tile column for this lane (4x16 f32 B: VGPR0 = rows K / K+2, VGPR1 = K+1 / K+3).
  const float* bcol = W + n0 + l15;

  v8f c = {};
  #pragma unroll 4
  for (int k = 0; k < IN_FEATS; k += 4) {
    const int ka = k + 2 * half;
    v2f a;
    a.x = arow[ka + 0];
    a.y = arow[ka + 1];
    v2f b;
    b.x = bcol[(ka + 0) * HIDDEN];
    b.y = bcol[(ka + 1) * HIDDEN];
    c = __builtin_amdgcn_wmma_f32_16x16x4_f32(
        /*neg_a=*/false, a, /*neg_b=*/false, b,
        /*c_mod=*/(short)0, c, /*reuse_a=*/false, /*reuse_b=*/false);
  }

  const float bias = bv[n0 + l15];
  // C/D layout: VGPR r -> row M = m0 + r + 8*half, col N = n0 + (lane&15)
  #pragma unroll
  for (int r = 0; r < 8; ++r) {
    const float v = c[r] + bias;
    const int idx = (m0 + r + 8 * half) * HIDDEN + n0 + l15;
    t_ws[idx] = v;
    out[idx]  = v;
  }
}

// ---------------------------------------------------------------------------
// Kernel B: for each edge e: out[dst[e]] += t[nbr[e]] + edge_feats[eid[e]].
// dst is sorted -> a wave's 16 consecutive edges mostly share dst, so we
// accumulate in registers and only flush atomics on segment boundaries.
// 256 floats/row across 32 lanes = 8 floats/lane = 2x float4 (b128 loads).
// ---------------------------------------------------------------------------
__device__ __forceinline__ void flush_row(float* __restrict__ out, int node,
                                          int lane, const float4& a0,
                                          const float4& a1) {
  float* p = out + node * HIDDEN;
  atomicAdd(p + 4 * lane + 0,       a0.x);
  atomicAdd(p + 4 * lane + 1,       a0.y);
  atomicAdd(p + 4 * lane + 2,       a0.z);
  atomicAdd(p + 4 * lane + 3,       a0.w);
  atomicAdd(p + 128 + 4 * lane + 0, a1.x);
  atomicAdd(p + 128 + 4 * lane + 1, a1.y);
  atomicAdd(p + 128 + 4 * lane + 2, a1.z);
  atomicAdd(p + 128 + 4 * lane + 3, a1.w);
}

__global__ __launch_bounds__(256) void edge_agg_kernel(
    const float* __restrict__ t,    // [N_NODES, HIDDEN] (workspace, read-only here)
    const float* __restrict__ ef,   // [N_EDGES, HIDDEN]
    const int* __restrict__ dst,
    const int* __restrict__ nbr,
    const int* __restrict__ eid,
    float* __restrict__ out)
{
  const int lane = threadIdx.x & 31;
  const int wave = (blockIdx.x * blockDim.x + threadIdx.x) >> 5;
  int e0 = wave * EDGES_PER_WAVE;
  if (e0 >= N_EDGES) return;
  int e1 = e0 + EDGES_PER_WAVE;
  if (e1 > N_EDGES) e1 = N_EDGES;

  float4 acc0 = make_float4(0.f, 0.f, 0.f, 0.f);
  float4 acc1 = make_float4(0.f, 0.f, 0.f, 0.f);
  int cur = dst[e0];

  for (int e = e0; e < e1; ++e) {
    const int d = dst[e];
    if (d != cur) {
      flush_row(out, cur, lane, acc0, acc1);
      acc0 = make_float4(0.f, 0.f, 0.f, 0.f);
      acc1 = make_float4(0.f, 0.f, 0.f, 0.f);
      cur = d;
    }
    const int nb = nbr[e];
    const int ei = eid[e];
    const float4* __restrict__ trow = (const float4*)(t + nb * HIDDEN);
    const float4* __restrict__ erow = (const float4*)(ef + ei * HIDDEN);
    const float4 t0 = trow[lane];
    const float4 t1 = trow[lane + 32];
    const float4 f0 = erow[lane];
    const float4 f1 = erow[lane + 32];
    acc0.x += t0.x + f0.x;  acc0.y += t0.y + f0.y;
    acc0.z += t0.z + f0.z;  acc0.w += t0.w + f0.w;
    acc1.x += t1.x + f1.x;  acc1.y += t1.y + f1.y;
    acc1.z += t1.z + f1.z;  acc1.w += t1.w + f1.w;
  }
  flush_row(out, cur, lane, acc0, acc1);
}

extern "C" void kernel_launch(void* const* d_in, const int* in_sizes, int n_in,
                              void* d_out, int out_size, void* d_ws, size_t ws_size,
                              hipStream_t stream) {
  const float* node_feats = (const float*)d_in[0];
  const float* edge_feats = (const float*)d_in[1];
  const float* W          = (const float*)d_in[2];
  const float* b          = (const float*)d_in[3];
  const int*   dst        = (const int*)d_in[4];
  const int*   nbr        = (const int*)d_in[5];
  const int*   eid        = (const int*)d_in[6];
  float* out = (float*)d_out;
  float* t   = (float*)d_ws;  // needs N_NODES*HIDDEN*4 = 51.2 MB

  // 3125 row tiles * 16 col tiles = 50000 waves, 8 waves per 256-thread block.
  const int gemmBlocks = (N_NODES / 16) * (HIDDEN / 16) / 8;  // 6250
  gemm_bias_wmma_kernel<<<gemmBlocks, 256, 0, stream>>>(node_feats, W, b, t, out);

  // 800000 edges / 16 per wave = 50000 waves -> 6250 blocks.
  const int edgeBlocks = (N_EDGES / EDGES_PER_WAVE + 7) / 8;  // 6250
  edge_agg_kernel<<<edgeBlocks, 256, 0, stream>>>(t, edge_feats, dst, nbr, eid, out);
}